// BiGraphContrastLayer_31353261260880
// MI455X (gfx1250) — compile-verified
//
#include <hip/hip_runtime.h>
#include <hip/hip_bf16.h>

#define DIN 128

typedef __attribute__((ext_vector_type(2))) float v2f;
typedef __attribute__((ext_vector_type(8))) float v8f;

// ---------------------------------------------------------------------------
// 1) Zero the output (used as agg accumulator) and the norm workspace.
// ---------------------------------------------------------------------------
__global__ void zero_kernel(float4* out4, long long n4, float* ws, long long nws) {
    long long stride = (long long)gridDim.x * blockDim.x;
    long long tid = (long long)blockIdx.x * blockDim.x + threadIdx.x;
    for (long long i = tid; i < n4; i += stride)
        out4[i] = make_float4(0.f, 0.f, 0.f, 0.f);
    for (long long i = tid; i < nws; i += stride)
        ws[i] = 0.f;
}

// ---------------------------------------------------------------------------
// 2) Degree histograms: ws[0..N)=deg_out_pos, [N..2N)=deg_in_pos,
//    [2N..3N)=deg_out_neg, [3N..4N)=deg_in_neg
// ---------------------------------------------------------------------------
__global__ void degree_kernel(const int* __restrict__ sp, const int* __restrict__ dp,
                              const int* __restrict__ sn, const int* __restrict__ dn,
                              float* __restrict__ ws, int N, int E) {
    int e = blockIdx.x * blockDim.x + threadIdx.x;
    if (e >= E) return;
    unsafeAtomicAdd(&ws[sp[e]], 1.f);
    unsafeAtomicAdd(&ws[(long long)N + dp[e]], 1.f);
    unsafeAtomicAdd(&ws[2LL * N + sn[e]], 1.f);
    unsafeAtomicAdd(&ws[3LL * N + dn[e]], 1.f);
}

// ---------------------------------------------------------------------------
// 3) deg -> norm = deg>0 ? rsqrt(deg) : 0   (in place, 4N elements)
// ---------------------------------------------------------------------------
__global__ void norm_kernel(float* ws, long long n) {
    long long i = (long long)blockIdx.x * blockDim.x + threadIdx.x;
    if (i >= n) return;
    float v = ws[i];
    ws[i] = (v > 0.f) ? rsqrtf(v) : 0.f;
}

// ---------------------------------------------------------------------------
// 4) SpMM scatter: agg[dst] += feats[src] * norm_s[src]
//    One edge per 32 lanes; each lane moves one float4 (128 feats total).
// ---------------------------------------------------------------------------
__global__ void __launch_bounds__(256)
scatter_kernel(const float* __restrict__ feats, const int* __restrict__ src,
               const int* __restrict__ dst, const float* __restrict__ norm_s,
               float* __restrict__ agg, int E) {
    int lane = threadIdx.x & 31;
    int eloc = threadIdx.x >> 5;
    long long e = (long long)blockIdx.x * (blockDim.x >> 5) + eloc;
    if (e >= E) return;
    int s = src[e];
    int d = dst[e];
    float ns = norm_s[s];
    float4 v = ((const float4*)(feats + (long long)s * DIN))[lane];
    float* op = agg + (long long)d * DIN + lane * 4;
    unsafeAtomicAdd(op + 0, v.x * ns);
    unsafeAtomicAdd(op + 1, v.y * ns);
    unsafeAtomicAdd(op + 2, v.z * ns);
    unsafeAtomicAdd(op + 3, v.w * ns);
}

// ---------------------------------------------------------------------------
// 5) In-place GEMM + norm_d + bias + PReLU via V_WMMA_F32_16X16X4_F32.
//
//    W is staged in LDS in K-PAIR-INTERLEAVED layout: element (k, n) lives at
//    Wp[(k>>1)*128 + col].{x:k even | y:k odd}, col = (n + ((kp&1)<<4)) & 127.
//    - The B fragment {W[k][n], W[k+1][n]} is then one aligned 8-byte LDS
//      word -> a single ds_load_b64 straight into the even VGPR pair the
//      WMMA B operand needs (no v_mov repacking).
//    - The +16-column swizzle on odd pair-rows shifts the upper half-wave's
//      dwords by 32 banks, so the wave's 64 dwords cover all 64 LDS banks.
//      On the read side the swizzle folds to col = (j*16 + lane) & 127.
// ---------------------------------------------------------------------------
__global__ void __launch_bounds__(256)
gemm_prelu_kernel(float* __restrict__ out, const float* __restrict__ W,
                  const float* __restrict__ bias, const float* __restrict__ prelu_a,
                  const float* __restrict__ norms, int N, int tilesPerGraph) {
    __shared__ float2 Wp[64 * 128];   // 64 KB

    // Cooperative stage + repack of W (coalesced global reads of rows 2kp, 2kp+1)
    for (int i = threadIdx.x; i < 64 * 128; i += blockDim.x) {
        int kp = i >> 7;                       // k-pair index 0..63
        int n  = i & 127;                      // column
        int col = (n + ((kp & 1) << 4)) & 127; // bank swizzle
        float2 w2;
        w2.x = W[(2 * kp) * DIN + n];
        w2.y = W[(2 * kp + 1) * DIN + n];
        Wp[kp * 128 + col] = w2;
    }
    __syncthreads();

    int wave = threadIdx.x >> 5;
    int lane = threadIdx.x & 31;
    long long gtile = (long long)blockIdx.x * (blockDim.x >> 5) + wave;
    long long totalTiles = 2LL * tilesPerGraph;
    if (gtile >= totalTiles) return;   // wave-uniform: EXEC stays all-ones for WMMA

    int g = (int)(gtile / tilesPerGraph);
    int t = (int)(gtile % tilesPerGraph);
    int m0 = t * 16;

    float* base = out + (long long)g * N * DIN;                 // agg in, h out (in place)
    const float* norm_d = norms + (long long)N * (1 + 2 * g);   // g=0 -> ws+N, g=1 -> ws+3N

    // A-fragment addressing (ISA 7.12.2, 32-bit A 16x4):
    //   lanes 0-15 : row m0+lane,    K = {k, k+1}   (even k-pair)
    //   lanes 16-31: row m0+lane-16, K = {k+2, k+3} (odd  k-pair)
    int hi = lane >> 4;                 // 0 for lanes 0-15, 1 for lanes 16-31
    int mrow = m0 + (lane & 15);
    if (mrow >= N) mrow = N - 1;        // tail clamp (N=100000 is 16-aligned)
    int koff = hi << 1;                 // 0 or 2

    const float* arow = base + (long long)mrow * DIN;
    float ndv = norm_d[mrow];           // fold norm_d into the A load

    v8f acc[8] = {};

    #pragma unroll 4
    for (int kk = 0; kk < DIN / 4; ++kk) {
        float2 av = *(const float2*)(arow + kk * 4 + koff);
        v2f a = {av.x * ndv, av.y * ndv};

        const float2* kb = Wp + (2 * kk + hi) * 128;  // this half-wave's k-pair row
        v2f b[8];
        #pragma unroll
        for (int j = 0; j < 8; ++j)
            b[j] = *(const v2f*)(kb + ((j * 16 + lane) & 127));

        #pragma unroll
        for (int j = 0; j < 8; ++j)
            acc[j] = __builtin_amdgcn_wmma_f32_16x16x4_f32(
                /*neg_a=*/false, a, /*neg_b=*/false, b[j],
                /*c_mod=*/(short)0, acc[j], /*reuse_a=*/false, /*reuse_b=*/false);
    }

    // Epilogue: bias + PReLU, store (C/D layout: VGPR v -> row v / v+8)
    float pa = prelu_a[0];
    int nlo = lane & 15;
    int mbase = m0 + hi * 8;

    if (m0 + 16 <= N) {
        // full tile: no per-row bounds checks
        #pragma unroll
        for (int j = 0; j < 8; ++j) {
            int c = j * 16 + nlo;
            float bv = bias[c];
            float* cp = base + (long long)mbase * DIN + c;
            #pragma unroll
            for (int v = 0; v < 8; ++v) {
                float h = acc[j][v] + bv;
                cp[(long long)v * DIN] = (h > 0.f) ? h : pa * h;
            }
        }
    } else {
        #pragma unroll
        for (int j = 0; j < 8; ++j) {
            int c = j * 16 + nlo;
            float bv = bias[c];
            #pragma unroll
            for (int v = 0; v < 8; ++v) {
                int r = mbase + v;
                if (r < N) {
                    float h = acc[j][v] + bv;
                    base[(long long)r * DIN + c] = (h > 0.f) ? h : pa * h;
                }
            }
        }
    }
}

// ---------------------------------------------------------------------------
// Launcher
// ---------------------------------------------------------------------------
extern "C" void kernel_launch(void* const* d_in, const int* in_sizes, int n_in,
                              void* d_out, int out_size, void* d_ws, size_t ws_size,
                              hipStream_t stream) {
    const float* feats   = (const float*)d_in[0];
    const float* W       = (const float*)d_in[1];
    const float* bias    = (const float*)d_in[2];
    const float* prelu_a = (const float*)d_in[3];
    const int* src_pos   = (const int*)d_in[4];
    const int* dst_pos   = (const int*)d_in[5];
    const int* src_neg   = (const int*)d_in[6];
    const int* dst_neg   = (const int*)d_in[7];

    float* out = (float*)d_out;
    float* ws  = (float*)d_ws;

    int N = in_sizes[0] / DIN;     // 100000
    int E = in_sizes[4];           // 800000

    long long n_out = 2LL * N * DIN;
    long long n4 = n_out / 4;
    long long nws = 4LL * N;

    // 1) zero agg accumulator (= d_out) and degree workspace
    zero_kernel<<<4096, 256, 0, stream>>>((float4*)out, n4, ws, nws);

    // 2) degrees
    degree_kernel<<<(E + 255) / 256, 256, 0, stream>>>(src_pos, dst_pos, src_neg,
                                                       dst_neg, ws, N, E);

    // 3) norms (in place over 4N floats)
    norm_kernel<<<(int)((nws + 255) / 256), 256, 0, stream>>>(ws, nws);

    // 4) SpMM scatter into d_out (agg), both graphs
    int sblocks = (E + 7) / 8;   // 8 edges per 256-thread block
    scatter_kernel<<<sblocks, 256, 0, stream>>>(feats, src_pos, dst_pos,
                                                ws /*norm_s_pos*/, out, E);
    scatter_kernel<<<sblocks, 256, 0, stream>>>(feats, src_neg, dst_neg,
                                                ws + 2LL * N /*norm_s_neg*/,
                                                out + (long long)N * DIN, E);

    // 5) in-place GEMM + norm_d + bias + PReLU (WMMA f32)
    int tilesPerGraph = (N + 15) / 16;
    long long totalWaves = 2LL * tilesPerGraph;
    int gblocks = (int)((totalWaves + 7) / 8);   // 8 waves per block
    gemm_prelu_kernel<<<gblocks, 256, 0, stream>>>(out, W, bias, prelu_a, ws, N,
                                                   tilesPerGraph);
}